// ScaledDotProductAttention_5231270167138
// MI455X (gfx1250) — compile-verified
//
#include <hip/hip_runtime.h>

// Shapes from the reference: B=2, H=16, S=2048, D=64 (fp32)
#define S_LEN 2048
#define D_DIM 64
#define BH    32              // B*H
#define NEGV  (-1.0e9f)
#define WAVES_PER_BLOCK 8
#define NCHUNK (S_LEN / 16)   // 128 key chunks of 16

typedef float v2f __attribute__((ext_vector_type(2)));
typedef float v4f __attribute__((ext_vector_type(4)));
typedef float v8f __attribute__((ext_vector_type(8)));
typedef unsigned int u32x4 __attribute__((ext_vector_type(4)));
typedef int i32x8 __attribute__((ext_vector_type(8)));
typedef int i32x4 __attribute__((ext_vector_type(4)));

__device__ __forceinline__ float groupmax16(float v) {
#pragma unroll
  for (int off = 1; off < 16; off <<= 1)
    v = fmaxf(v, __shfl_xor(v, off, 32));
  return v;
}

__device__ __forceinline__ float groupsum16(float v) {
#pragma unroll
  for (int off = 1; off < 16; off <<= 1)
    v += __shfl_xor(v, off, 32);
  return v;
}

// Tensor Data Mover: DMA a 16-row x 64-col fp32 tile (row stride 64 floats)
// from global memory into LDS. Descriptor per CDNA5 ISA.
// Tracked by TENSORcnt; this toolchain uses the 6-arg builtin form
// (uint32x4 g0, int32x8 g1, int32x4 g2, int32x4 g3, int32x8 pad, i32 cpol).
__device__ __forceinline__ void tdm_load_tile16x64(unsigned ldsByteAddr,
                                                   const float* gsrc) {
  unsigned long long ga = (unsigned long long)(const void*)gsrc;
  u32x4 g0;
  g0[0] = 1u;                                            // count=1 (valid user D#)
  g0[1] = ldsByteAddr;                                   // lds_addr [63:32]
  g0[2] = (unsigned)(ga & 0xFFFFFFFFull);                // global_addr[31:0]
  g0[3] = (unsigned)((ga >> 32) & 0x01FFFFFFull)         // global_addr[56:32]
        | 0x80000000u;                                   // type=2 ("image")
  i32x8 g1;
  g1[0] = 0x00020000;        // data_size=2 (4 bytes), workgroup_mask=0
  g1[1] = (64 << 16);        // tensor_dim0 = 64  (bits [79:48], low 16 here)
  g1[2] = (int)(2048u << 16);// tensor_dim1 = 2048 (bits [111:80], low 16 here)
  g1[3] = (64 << 16);        // tile_dim0 = 64   (bits [127:112])
  g1[4] = 16;                // tile_dim1 = 16   (bits [143:128]); tile_dim2=0
  g1[5] = 64;                // tensor_dim0_stride = 64 (bits [207:160], low 32)
  g1[6] = 0;                 // dim0_stride hi / dim1_stride lo
  g1[7] = 0;
  i32x4 z4 = {0, 0, 0, 0};           // groups 2/3 unused (2D tensor)
  i32x8 z8 = {0, 0, 0, 0, 0, 0, 0, 0};
  __builtin_amdgcn_tensor_load_to_lds(g0, g1, z4, z4, z8, 0);
}

// One wave32 handles a 16-query-row tile of one (b,h) head; the 8 waves of a
// block cover 8 consecutive q-tiles of the SAME head, so K/V tiles are staged
// once per block into LDS by the TDM (double-buffered) and shared by all waves.
__global__ __launch_bounds__(WAVES_PER_BLOCK * 32)
void fa_fp32_wmma_kernel(const float* __restrict__ Q,
                         const float* __restrict__ K,
                         const float* __restrict__ V,
                         const unsigned char* __restrict__ mask,
                         float* __restrict__ prob,
                         float* __restrict__ attn) {
  __shared__ float shK[2][16 * D_DIM];               // TDM-staged K tiles (2x4KB)
  __shared__ float shV[2][16 * D_DIM];               // TDM-staged V tiles (2x4KB)
  __shared__ float shP[WAVES_PER_BLOCK][16 * 20];    // per-wave P-tile transpose

  const int lane  = threadIdx.x & 31;
  const int wave  = threadIdx.x >> 5;
  const int wtile = blockIdx.x * WAVES_PER_BLOCK + wave;  // 0..4095
  const int bh    = wtile >> 7;            // / (S/16); uniform across block
  const int q0    = (wtile & 127) << 4;    // query-row base
  const int n     = lane & 15;             // column-in-tile / row-in-A
  const int h     = lane >> 4;             // lane half

  const float* Qb = Q + (size_t)bh * S_LEN * D_DIM + (size_t)q0 * D_DIM;
  const float* Kb = K + (size_t)bh * S_LEN * D_DIM;
  const float* Vb = V + (size_t)bh * S_LEN * D_DIM;
  const unsigned char* Mb = mask + (size_t)bh * S_LEN * S_LEN + (size_t)q0 * S_LEN;
  float* Ab = attn + (size_t)bh * S_LEN * S_LEN + (size_t)q0 * S_LEN;
  float* Pb = prob + (size_t)bh * S_LEN * D_DIM + (size_t)q0 * D_DIM;

  const unsigned ldsK0 = (unsigned)(unsigned long long)(const void*)&shK[0][0];
  const unsigned ldsK1 = (unsigned)(unsigned long long)(const void*)&shK[1][0];
  const unsigned ldsV0 = (unsigned)(unsigned long long)(const void*)&shV[0][0];
  const unsigned ldsV1 = (unsigned)(unsigned long long)(const void*)&shV[1][0];

  // ---- Q tile (16x64) in WMMA f32 A-layout, pre-scaled by 1/sqrt(D) ----
  // A 16x4 layout: lane L -> row M=L%16; VGPR0 holds K=2h, VGPR1 holds K=2h+1.
  v2f qa[16];
#pragma unroll
  for (int kc = 0; kc < 16; ++kc) {
    const float* p = Qb + n * D_DIM + 4 * kc + 2 * h;
    qa[kc].x = p[0] * 0.125f;
    qa[kc].y = p[1] * 0.125f;
  }

  // Per-lane online softmax state: this lane owns column (chunk-local n) of
  // row r+8h; cross-lane merge happens once after the chunk loop (no per-chunk
  // DS-pipe butterflies -> DS pipe stays free for tile loads).
  float m[8], l[8];
#pragma unroll
  for (int r = 0; r < 8; ++r) { m[r] = -3.0e38f; l[r] = 0.0f; }

  // ======================= Pass 1: softmax stats =======================
  if (wave == 0) tdm_load_tile16x64(ldsK0, Kb);
  for (int j = 0; j < NCHUNK; ++j) {
    const int kb = j << 4;
    if (wave == 0) {
      if (j + 1 < NCHUNK) {
        tdm_load_tile16x64((j & 1) ? ldsK0 : ldsK1,
                           Kb + (size_t)(kb + 16) * D_DIM);
        __builtin_amdgcn_s_wait_tensorcnt(1);   // current tile landed
      } else {
        __builtin_amdgcn_s_wait_tensorcnt(0);
      }
    }
    __syncthreads();                            // tile visible to all waves

    const float* kt = shK[j & 1] + n * D_DIM + 2 * h;
    v8f c0 = {}, c1 = {};                       // two chains for WMMA ILP
#pragma unroll
    for (int kc = 0; kc < 16; kc += 2) {
      const v2f bK0 = *(const v2f*)(kt + 4 * kc);      // B-layout: kd = 2h,2h+1
      const v2f bK1 = *(const v2f*)(kt + 4 * kc + 4);
      c0 = __builtin_amdgcn_wmma_f32_16x16x4_f32(false, qa[kc], false, bK0,
                                                 (short)0, c0, false, false);
      c1 = __builtin_amdgcn_wmma_f32_16x16x4_f32(false, qa[kc + 1], false, bK1,
                                                 (short)0, c1, false, false);
    }
    const v8f c = c0 + c1;
#pragma unroll
    for (int r = 0; r < 8; ++r) {
      float sc = c[r];
      if (Mb[(size_t)(r + 8 * h) * S_LEN + kb + n]) sc = NEGV;
      const float mn = fmaxf(m[r], sc);
      l[r] = l[r] * __expf(m[r] - mn) + __expf(sc - mn);
      m[r] = mn;
    }
    __syncthreads();                            // done reading before overwrite
  }

  // Cross-lane merge of the 16 per-column partials (once, not per chunk).
  float rl[8];
#pragma unroll
  for (int r = 0; r < 8; ++r) {
    const float mg = groupmax16(m[r]);
    l[r]  = groupsum16(l[r] * __expf(m[r] - mg));
    m[r]  = mg;
    rl[r] = 1.0f / l[r];
  }

  v8f o[4];
#pragma unroll
  for (int g = 0; g < 4; ++g) o[g] = (v8f){};

  float* myP = shP[wave];

  // ============ Pass 2: emit attn + accumulate prob = P @ V ============
  if (wave == 0) {
    tdm_load_tile16x64(ldsK0, Kb);
    tdm_load_tile16x64(ldsV0, Vb);
  }
  for (int j = 0; j < NCHUNK; ++j) {
    const int kb = j << 4;
    if (wave == 0) {
      if (j + 1 < NCHUNK) {
        tdm_load_tile16x64((j & 1) ? ldsK0 : ldsK1,
                           Kb + (size_t)(kb + 16) * D_DIM);
        tdm_load_tile16x64((j & 1) ? ldsV0 : ldsV1,
                           Vb + (size_t)(kb + 16) * D_DIM);
        __builtin_amdgcn_s_wait_tensorcnt(2);   // current K+V landed
      } else {
        __builtin_amdgcn_s_wait_tensorcnt(0);
      }
    }
    __syncthreads();

    const float* kt = shK[j & 1] + n * D_DIM + 2 * h;
    const float* vt = shV[j & 1];
    v8f c0 = {}, c1 = {};
#pragma unroll
    for (int kc = 0; kc < 16; kc += 2) {
      const v2f bK0 = *(const v2f*)(kt + 4 * kc);
      const v2f bK1 = *(const v2f*)(kt + 4 * kc + 4);
      c0 = __builtin_amdgcn_wmma_f32_16x16x4_f32(false, qa[kc], false, bK0,
                                                 (short)0, c0, false, false);
      c1 = __builtin_amdgcn_wmma_f32_16x16x4_f32(false, qa[kc + 1], false, bK1,
                                                 (short)0, c1, false, false);
    }
    const v8f c = c0 + c1;

    // Normalized probabilities into the per-wave LDS tile (stride 20 keeps
    // float4 alignment and spreads banks: gcd(20,64)=4 -> 16 distinct banks).
#pragma unroll
    for (int r = 0; r < 8; ++r) {
      float sc = c[r];
      if (Mb[(size_t)(r + 8 * h) * S_LEN + kb + n]) sc = NEGV;
      myP[(r + 8 * h) * 20 + n] = __expf(sc - m[r]) * rl[r];
    }
    asm volatile("s_wait_dscnt 0" ::: "memory");  // within-wave LDS ordering

    // Wide attn writeback: lane -> row n, cols kb+8h..kb+8h+7 (2x b128).
    {
      const v4f lo = *(const v4f*)&myP[n * 20 + 8 * h];
      const v4f hi = *(const v4f*)&myP[n * 20 + 8 * h + 4];
      float* ap = Ab + (size_t)n * S_LEN + kb + 8 * h;
      *(v4f*)ap       = lo;
      *(v4f*)(ap + 4) = hi;
    }

    // P tile (16x16) in A-layout: lane L -> row n, K = 4*kc + 2h + {0,1}
    v2f pa[4];
#pragma unroll
    for (int kc = 0; kc < 4; ++kc) {
      const int col = 4 * kc + 2 * h;
      pa[kc].x = myP[n * 20 + col];
      pa[kc].y = myP[n * 20 + col + 1];
    }

    // o[g] (16 rows x d-cols 16g..16g+15) += P(16x16) @ V(16x16 chunk)
#pragma unroll
    for (int g = 0; g < 4; ++g) {
#pragma unroll
      for (int kc = 0; kc < 4; ++kc) {
        v2f bV;
        bV.x = vt[(4 * kc + 2 * h) * D_DIM + 16 * g + n];
        bV.y = vt[(4 * kc + 2 * h + 1) * D_DIM + 16 * g + n];
        o[g] = __builtin_amdgcn_wmma_f32_16x16x4_f32(false, pa[kc], false, bV,
                                                     (short)0, o[g], false, false);
      }
    }
    __syncthreads();                            // done reading before overwrite
  }

  // ---- Write prob tile (16x64) through LDS for b128 stores ----
#pragma unroll
  for (int g = 0; g < 4; ++g) {
#pragma unroll
    for (int r = 0; r < 8; ++r)
      myP[(r + 8 * h) * 20 + n] = o[g][r];
    asm volatile("s_wait_dscnt 0" ::: "memory");
    const v4f lo = *(const v4f*)&myP[n * 20 + 8 * h];
    const v4f hi = *(const v4f*)&myP[n * 20 + 8 * h + 4];
    float* pp = Pb + (size_t)n * D_DIM + 16 * g + 8 * h;
    *(v4f*)pp       = lo;
    *(v4f*)(pp + 4) = hi;
    asm volatile("s_wait_dscnt 0" ::: "memory");  // loads done before next overwrite
  }
}

extern "C" void kernel_launch(void* const* d_in, const int* in_sizes, int n_in,
                              void* d_out, int out_size, void* d_ws, size_t ws_size,
                              hipStream_t stream) {
  const float* Q = (const float*)d_in[0];
  const float* K = (const float*)d_in[1];
  const float* V = (const float*)d_in[2];
  const unsigned char* mask = (const unsigned char*)d_in[3];

  float* out  = (float*)d_out;
  float* prob = out;                                   // B*H*S*D floats
  float* attn = out + (size_t)BH * S_LEN * D_DIM;      // B*H*S*S floats

  const int tiles = BH * (S_LEN / 16);                 // 4096 wave tiles
  dim3 grid(tiles / WAVES_PER_BLOCK);
  dim3 block(WAVES_PER_BLOCK * 32);
  hipLaunchKernelGGL(fa_fp32_wmma_kernel, grid, block, 0, stream,
                     Q, K, V, mask, prob, attn);
}